// PatchPooler_58351425683690
// MI455X (gfx1250) — compile-verified
//
#include <hip/hip_runtime.h>
#include <hip/hip_bf16.h>

// Problem constants (fixed by the reference).
constexpr int Bn = 8;
constexpr int Sn = 4096;
constexpr int Dn = 1024;
constexpr int Pn = 512;

typedef float v2f __attribute__((ext_vector_type(2)));
typedef float v8f __attribute__((ext_vector_type(8)));

// ---------------------------------------------------------------------------
// Kernel 1: per batch row, turn boundary bits into segment start positions.
// starts[b][p] = token index of p-th boundary; starts[b][P] = S.
// One 256-thread block per row; each thread scans 16 consecutive tokens,
// block-wide Hillis-Steele scan over the 256 partial counts in LDS.
// ---------------------------------------------------------------------------
__global__ __launch_bounds__(256) void seg_starts_kernel(
    const int* __restrict__ bnd, int* __restrict__ starts) {
  const int b = blockIdx.x;
  const int tid = threadIdx.x;
  const int base = tid * 16;
  const int* row = bnd + (size_t)b * Sn;

  __shared__ int psum[256];

  int vals[16];
  int local = 0;
#pragma unroll
  for (int i = 0; i < 16; ++i) {
    vals[i] = row[base + i];
    local += vals[i];
  }
  psum[tid] = local;
  __syncthreads();

  // Inclusive scan over the 256 partials.
  for (int off = 1; off < 256; off <<= 1) {
    int add = (tid >= off) ? psum[tid - off] : 0;
    __syncthreads();
    psum[tid] += add;
    __syncthreads();
  }

  int running = psum[tid] - local;  // exclusive prefix = ones before my chunk
  int* srow = starts + (size_t)b * (Pn + 1);
#pragma unroll
  for (int i = 0; i < 16; ++i) {
    if (vals[i]) {
      srow[running] = base + i;  // seg id of a boundary token == ones-before
      ++running;
    }
  }
  if (tid == 0) srow[Pn] = Sn;
}

// ---------------------------------------------------------------------------
// Kernel 2: WMMA segment mean pooling.
// Each workgroup: (batch b, block of 16 segments p0..p0+15, 256-channel slab).
// 128 threads = 4 waves; wave w owns channels [slab + 64*w, +64) as four
// 16-wide N-tiles. K loop runs over the token span of the 16 segments in
// chunks of 4, using V_WMMA_F32_16X16X4_F32:
//   A (16x4, MxK): lane = M (lane&15); lane-half picks K pair {0,1}/{2,3};
//                  A.x = K=2*hi, A.y = K=2*hi+1. One-hot selection built from
//                  per-lane segment bounds (two compares, no memory).
//   B (4x16, KxN): lane = N (lane&15); lane-half picks K pair; VGPR0 = K=2*hi,
//                  VGPR1 = K=2*hi+1. Loaded straight from x (row-major in d).
//   C/D (16x16):   VGPR j holds row M = j + 8*hi, N = lane&15.
// ---------------------------------------------------------------------------
__global__ __launch_bounds__(128) void pool_wmma_kernel(
    const float* __restrict__ x, const int* __restrict__ starts,
    float* __restrict__ out) {
  constexpr int DCHUNK = 256;            // channels per workgroup
  constexpr int NPB = Pn / 16;           // 32 segment-blocks per batch row
  constexpr int ND = Dn / DCHUNK;        // 4 channel slabs

  const int wg = blockIdx.x;
  const int dchunk = wg % ND;
  const int pb = (wg / ND) % NPB;
  const int b = wg / (ND * NPB);
  const int p0 = pb * 16;

  __shared__ int sStarts[17];
  const int tid = threadIdx.x;
  if (tid < 17) sStarts[tid] = starts[(size_t)b * (Pn + 1) + p0 + tid];
  __syncthreads();

  const int lane = tid & 31;
  const int wave = tid >> 5;             // 0..3
  const int hi = lane >> 4;              // lane-half: K pair selector
  const int mrow = lane & 15;            // A-matrix row (segment within block)
  const int n = lane & 15;               // B/C column (channel within tile)

  const int startM = sStarts[mrow];      // my segment's [start, end)
  const int endM = sStarts[mrow + 1];
  const int s0 = sStarts[0];             // token span of the 16 segments
  const int s1 = sStarts[16];

  const int dbase = dchunk * DCHUNK + wave * 64;
  const float* xb = x + (size_t)b * Sn * Dn;

  v8f c0 = {}, c1 = {}, c2 = {}, c3 = {};
  const int koff = hi * 2;

  for (int s = s0; s < s1; s += 4) {
    const int ka = s + koff;             // token index for A.x / B VGPR0
    const int kb = ka + 1;               // token index for A.y / B VGPR1

    v2f A;
    A.x = (ka >= startM && ka < endM) ? 1.0f : 0.0f;
    A.y = (kb >= startM && kb < endM) ? 1.0f : 0.0f;

    // Clamp tail token addresses (their A column is all-zero anyway).
    const int kac = (ka < Sn - 1) ? ka : (Sn - 1);
    const int kbc = (kb < Sn - 1) ? kb : (Sn - 1);
    const float* ra = xb + (size_t)kac * Dn + dbase + n;
    const float* rb = xb + (size_t)kbc * Dn + dbase + n;

    v2f B0, B1, B2, B3;
    B0.x = ra[0];  B0.y = rb[0];
    B1.x = ra[16]; B1.y = rb[16];
    B2.x = ra[32]; B2.y = rb[32];
    B3.x = ra[48]; B3.y = rb[48];

    c0 = __builtin_amdgcn_wmma_f32_16x16x4_f32(false, A, false, B0, (short)0,
                                               c0, false, false);
    c1 = __builtin_amdgcn_wmma_f32_16x16x4_f32(false, A, false, B1, (short)0,
                                               c1, false, false);
    c2 = __builtin_amdgcn_wmma_f32_16x16x4_f32(false, A, false, B2, (short)0,
                                               c2, false, false);
    c3 = __builtin_amdgcn_wmma_f32_16x16x4_f32(false, A, false, B3, (short)0,
                                               c3, false, false);
  }

  // Scale by 1/count and store. VGPR j of C/D holds row M = j + 8*hi.
#pragma unroll
  for (int j = 0; j < 8; ++j) {
    const int rl = j + hi * 8;
    const float inv = 1.0f / (float)(sStarts[rl + 1] - sStarts[rl]);
    const size_t o = ((size_t)(b * Pn + p0 + rl)) * Dn + dbase + n;
    out[o + 0]  = c0[j] * inv;
    out[o + 16] = c1[j] * inv;
    out[o + 32] = c2[j] * inv;
    out[o + 48] = c3[j] * inv;
  }
}

extern "C" void kernel_launch(void* const* d_in, const int* in_sizes, int n_in,
                              void* d_out, int out_size, void* d_ws,
                              size_t ws_size, hipStream_t stream) {
  const float* x = (const float*)d_in[0];       // [B, S, D] f32
  const int* boundaries = (const int*)d_in[1];  // [B, S] i32
  // d_in[2] = num_patches scalar (== Pn), fixed by the reference.
  float* out = (float*)d_out;                   // [B, P, D] f32

  int* starts = (int*)d_ws;                     // B * (P+1) ints of scratch

  seg_starts_kernel<<<Bn, 256, 0, stream>>>(boundaries, starts);

  constexpr int DCHUNK = 256;
  const int grid = Bn * (Pn / 16) * (Dn / DCHUNK);  // 1024 workgroups
  pool_wmma_kernel<<<grid, 128, 0, stream>>>(x, starts, out);
}